// SeparateAudioTokenizer_39943195853392
// MI455X (gfx1250) — compile-verified
//
#include <hip/hip_runtime.h>

typedef __attribute__((ext_vector_type(2))) float v2f;
typedef __attribute__((ext_vector_type(8))) float v8f;

#define VQ_D 8192   // embedding dim (GEMM K)
#define VQ_K 4096   // codebook entries (GEMM N)
#define VQ_M 4096   // B*C rows (GEMM M)
#define SA   18     // padded tile row stride in v2f (144B: B128-aligned, bank-spread)

// -------- CDNA5 async global->LDS copy (ASYNCcnt) ------------------------------
__device__ __forceinline__ void async_b128(unsigned lds_off, const float* gp) {
  unsigned long long ga = (unsigned long long)(const void*)gp;
  asm volatile("global_load_async_to_lds_b128 %0, %1, off"
               :: "v"(lds_off), "v"(ga) : "memory");
}
__device__ __forceinline__ unsigned lds_off32(const void* p) {
  // generic pointers to LDS carry the wave-relative LDS offset in the low 32 bits
  return (unsigned)(size_t)p;
}
#define WAIT_ASYNC_LE8() asm volatile("s_wait_asynccnt 0x8" ::: "memory")
#define WAIT_ASYNC_0()   asm volatile("s_wait_asynccnt 0x0" ::: "memory")

// ---------------------------------------------------------------- conv layer 0 (Cin=1)
__global__ void conv0_kernel(const float* __restrict__ x, const float* __restrict__ w,
                             const float* __restrict__ bias, float* __restrict__ y,
                             int Tin, int Tout) {
  int t  = blockIdx.x * blockDim.x + threadIdx.x;
  int co = blockIdx.y;
  int b  = blockIdx.z;
  const float* xr = x + (size_t)b * Tin;
  float acc = bias[co];
  int base = 2 * t - 2;
#pragma unroll
  for (int k = 0; k < 5; ++k) {
    int tin = base + k;
    float xv = (tin >= 0 && tin < Tin) ? xr[tin] : 0.0f;
    acc = fmaf(w[co * 5 + k], xv, acc);
  }
  y[((size_t)b * gridDim.y + co) * Tout + t] = fmaxf(acc, 0.0f);
}

// ---------------------------------------------------------------- conv layers 1..3:
// implicit-GEMM on fp32 WMMA. M=Cout (tile 128), N=128 output t's of one batch,
// K=CIN*5 in chunks of 40 (8 input channels). Weights [Cout][CIN*5] are already
// the row-major A matrix; the stride-2 conv indexing happens in the B-frag fetch.
template <int CIN>
__global__ void __launch_bounds__(256) convw_kernel(
    const float* __restrict__ x, const float* __restrict__ w,
    const float* __restrict__ bias, float* __restrict__ y,
    int Tin, int Tout, int Cout) {
  constexpr int KTOT = CIN * 5;
  constexpr int SPX  = 264;                 // padded x row stride (260 used)
  __shared__ v2f   wsh[128 * 21];           // 128 co x 20 k-pairs (stride 21)
  __shared__ float xs[8 * SPX];             // 8 ci x 260 input samples

  const int tid = threadIdx.x;
  const int lane = tid & 31;
  const int wv = tid >> 5, wm = wv & 1, wn = wv >> 1;
  const int hi = lane >> 4, l15 = lane & 15;
  const int t0  = blockIdx.x * 128;
  const int co0 = blockIdx.y * 128;
  const int b   = blockIdx.z;
  const int tin0 = 2 * t0 - 2;
  const float* xb = x + (size_t)b * CIN * Tin;

  const v8f zero8 = {0.f, 0.f, 0.f, 0.f, 0.f, 0.f, 0.f, 0.f};
  v8f acc[4][2];
#pragma unroll
  for (int mi = 0; mi < 4; ++mi)
#pragma unroll
    for (int ni = 0; ni < 2; ++ni) acc[mi][ni] = zero8;

  for (int kc = 0; kc < KTOT; kc += 40) {
    const int ci0 = kc / 5;
    __syncthreads();
    for (int i = tid; i < 128 * 20; i += 256) {        // stage A (weights)
      int r = i / 20, j = i - r * 20;
      const float* wp = w + (size_t)(co0 + r) * KTOT + kc + 2 * j;
      v2f p; p.x = wp[0]; p.y = wp[1];
      wsh[r * 21 + j] = p;
    }
    for (int i = tid; i < 8 * 260; i += 256) {         // stage x tile (with halo)
      int ci = i / 260, j = i - ci * 260;
      int tin = tin0 + j;
      xs[ci * SPX + j] = (tin >= 0 && tin < Tin)
                             ? xb[(size_t)(ci0 + ci) * Tin + tin] : 0.0f;
    }
    __syncthreads();
#pragma unroll
    for (int kk = 0; kk < 40; kk += 4) {
      const int kd0 = kk + 2 * hi;                     // even
      const int ciA = kd0 / 5, kqA = kd0 - 5 * ciA;
      const int kd1 = kd0 + 1;
      const int ciB = kd1 / 5, kqB = kd1 - 5 * ciB;
      v2f a[4], bfr[2];
#pragma unroll
      for (int mi = 0; mi < 4; ++mi)
        a[mi] = wsh[(wm * 64 + mi * 16 + l15) * 21 + (kd0 >> 1)];
#pragma unroll
      for (int ni = 0; ni < 2; ++ni) {
        const int tl = 2 * (wn * 32 + ni * 16 + l15);
        v2f p;
        p.x = xs[ciA * SPX + tl + kqA];
        p.y = xs[ciB * SPX + tl + kqB];
        bfr[ni] = p;
      }
#pragma unroll
      for (int mi = 0; mi < 4; ++mi)
#pragma unroll
        for (int ni = 0; ni < 2; ++ni)
          acc[mi][ni] = __builtin_amdgcn_wmma_f32_16x16x4_f32(
              false, a[mi], false, bfr[ni], (short)0, acc[mi][ni], false, false);
    }
  }
  // epilogue: bias + ReLU, scatter per C/D fragment layout
#pragma unroll
  for (int mi = 0; mi < 4; ++mi)
#pragma unroll
    for (int v = 0; v < 8; ++v) {
      const int row = wm * 64 + mi * 16 + hi * 8 + v;
      const int co = co0 + row;
      const float bv = bias[co];
#pragma unroll
      for (int ni = 0; ni < 2; ++ni) {
        const int t = t0 + wn * 32 + ni * 16 + l15;
        y[((size_t)b * Cout + co) * Tout + t] = fmaxf(acc[mi][ni][v] + bv, 0.0f);
      }
    }
}

// ---------------------------------------------------------------- 0.5*||e_k||^2
__global__ void enorm_kernel(const float* __restrict__ cb, float* __restrict__ enorm) {
  __shared__ float red[256];
  const float* c = cb + (size_t)blockIdx.x * VQ_D;
  float s = 0.f;
  for (int i = threadIdx.x; i < VQ_D; i += 256) { float v = c[i]; s = fmaf(v, v, s); }
  red[threadIdx.x] = s;
  __syncthreads();
  for (int off = 128; off > 0; off >>= 1) {
    if (threadIdx.x < off) red[threadIdx.x] += red[threadIdx.x + off];
    __syncthreads();
  }
  if (threadIdx.x == 0) enorm[blockIdx.x] = 0.5f * red[0];
}

// ---------------------------------------------------------------- fused VQ:
// async-pipelined fp32 WMMA GEMM (x . e^T) -> score -> per-row argmax via packed
// u64 LDS atomic max -> gather codebook rows (straight-through forward == q).
__device__ __forceinline__ unsigned fkey(float f) {
  unsigned u = __float_as_uint(f);
  return (u & 0x80000000u) ? ~u : (u | 0x80000000u);  // order-preserving f32->u32
}

__global__ void __launch_bounds__(256) vq_kernel(const float* __restrict__ z,
                                                 const float* __restrict__ cb,
                                                 const float* __restrict__ enorm,
                                                 float* __restrict__ out) {
  __shared__ v2f tA[2][128 * SA];   // x rows    [m][k-pair], double buffered
  __shared__ v2f tB[2][128 * SA];   // codebook  [n][k-pair], double buffered
  __shared__ unsigned long long best[128];
  __shared__ int idxs[128];

  const int tid = threadIdx.x;
  const int lane = tid & 31;
  const int wv = tid >> 5, wm = wv & 1, wn = wv >> 1;
  const int hi = lane >> 4, l15 = lane & 15;
  const int m0 = blockIdx.x * 128;

  if (tid < 128) best[tid] = 0ull;

  const v8f zero8 = {0.f, 0.f, 0.f, 0.f, 0.f, 0.f, 0.f, 0.f};

  // per-thread tile-element coordinates (2 B128 per tile per chunk)
  const int em = tid >> 1;                 // row 0..127 (2 threads per row)
  const int eq = (tid & 1) * 4;            // k-pair 0/4 base, +2 inner

  for (int n0 = 0; n0 < VQ_K; n0 += 128) {
    v8f acc[4][2];
#pragma unroll
    for (int mi = 0; mi < 4; ++mi)
#pragma unroll
      for (int ni = 0; ni < 2; ++ni) acc[mi][ni] = zero8;

    // prologue: async-load chunk 0 into buffer 0
#pragma unroll
    for (int h = 0; h < 2; ++h) {
      async_b128(lds_off32(&tA[0][em * SA + eq + 2 * h]),
                 z + (size_t)(m0 + em) * VQ_D + 4 * (eq + 2 * h));
      async_b128(lds_off32(&tB[0][em * SA + eq + 2 * h]),
                 cb + (size_t)(n0 + em) * VQ_D + 4 * (eq + 2 * h));
    }

    for (int c = 0; c < VQ_D / 32; ++c) {
      const int cur = c & 1;
      if (c + 1 < VQ_D / 32) {             // prefetch next chunk into other buffer
        const int kn = 32 * (c + 1);
#pragma unroll
        for (int h = 0; h < 2; ++h) {
          async_b128(lds_off32(&tA[cur ^ 1][em * SA + eq + 2 * h]),
                     z + (size_t)(m0 + em) * VQ_D + kn + 4 * (eq + 2 * h));
          async_b128(lds_off32(&tB[cur ^ 1][em * SA + eq + 2 * h]),
                     cb + (size_t)(n0 + em) * VQ_D + kn + 4 * (eq + 2 * h));
        }
        WAIT_ASYNC_LE8();                  // <=8 pending: chunk c's 4 copies may still be in flight
      } else {
        WAIT_ASYNC_0();
      }
      __syncthreads();
#pragma unroll
      for (int kk = 0; kk < 32; kk += 4) {
        const int k2 = (kk >> 1) + hi;     // lane-half selects the ISA K pair
        v2f a[4], bfr[2];
#pragma unroll
        for (int mi = 0; mi < 4; ++mi)
          a[mi] = tA[cur][(wm * 64 + mi * 16 + l15) * SA + k2];
#pragma unroll
        for (int ni = 0; ni < 2; ++ni)
          bfr[ni] = tB[cur][(wn * 32 + ni * 16 + l15) * SA + k2];
#pragma unroll
        for (int mi = 0; mi < 4; ++mi)
#pragma unroll
          for (int ni = 0; ni < 2; ++ni)
            acc[mi][ni] = __builtin_amdgcn_wmma_f32_16x16x4_f32(
                false, a[mi], false, bfr[ni], (short)0, acc[mi][ni], false, false);
      }
      __syncthreads();
    }

    // fold this N-tile's scores into the per-row running best
#pragma unroll
    for (int ni = 0; ni < 2; ++ni) {
      const int ncol = n0 + wn * 32 + ni * 16 + l15;
      const float en = enorm[ncol];
#pragma unroll
      for (int mi = 0; mi < 4; ++mi)
#pragma unroll
        for (int v = 0; v < 8; ++v) {
          const int row = wm * 64 + mi * 16 + hi * 8 + v;
          const float s = acc[mi][ni][v] - en;
          unsigned long long packed =
              ((unsigned long long)fkey(s) << 32) | (unsigned)ncol;
          atomicMax(&best[row], packed);
        }
    }
  }
  __syncthreads();
  if (tid < 128) idxs[tid] = (int)(best[tid] & 0xffffffffu);
  __syncthreads();

  for (int r = 0; r < 128; ++r) {
    const float4* src = (const float4*)(cb + (size_t)idxs[r] * VQ_D);
    float4* dst = (float4*)(out + (size_t)(m0 + r) * VQ_D);
    for (int i = tid; i < VQ_D / 4; i += 256) dst[i] = src[i];
  }
}

// ---------------------------------------------------------------- launch
extern "C" void kernel_launch(void* const* d_in, const int* in_sizes, int n_in,
                              void* d_out, int out_size, void* d_ws, size_t ws_size,
                              hipStream_t stream) {
  const float* vocal = (const float*)d_in[0];
  const float* bgm   = (const float*)d_in[1];
  const float* vw[4] = {(const float*)d_in[2], (const float*)d_in[4],
                        (const float*)d_in[6], (const float*)d_in[8]};
  const float* vb[4] = {(const float*)d_in[3], (const float*)d_in[5],
                        (const float*)d_in[7], (const float*)d_in[9]};
  const float* bw[4] = {(const float*)d_in[10], (const float*)d_in[12],
                        (const float*)d_in[14], (const float*)d_in[16]};
  const float* bb[4] = {(const float*)d_in[11], (const float*)d_in[13],
                        (const float*)d_in[15], (const float*)d_in[17]};
  const float* cb = (const float*)d_in[18];
  float* out = (float*)d_out;

  const size_t BUF = 33554432;            // 128 MiB of floats
  float* buf0  = (float*)d_ws;
  float* buf1  = buf0 + BUF;
  float* enorm = buf1 + BUF;              // 4096 floats

  enorm_kernel<<<4096, 256, 0, stream>>>(cb, enorm);

  for (int e = 0; e < 2; ++e) {
    const float* inx = e ? bgm : vocal;
    const float* const* W  = e ? bw : vw;
    const float* const* Bb = e ? bb : vb;
    // L0: 1->64, 131072->65536 (tiny; VALU)
    conv0_kernel<<<dim3(65536 / 256, 64, 8), 256, 0, stream>>>(
        inx, W[0], Bb[0], buf0, 131072, 65536);
    // L1..L3: implicit-GEMM fp32 WMMA
    convw_kernel<64><<<dim3(32768 / 128, 1, 8), 256, 0, stream>>>(
        buf0, W[1], Bb[1], buf1, 65536, 32768, 128);
    convw_kernel<128><<<dim3(16384 / 128, 2, 8), 256, 0, stream>>>(
        buf1, W[2], Bb[2], buf0, 32768, 16384, 256);
    convw_kernel<256><<<dim3(8192 / 128, 4, 8), 256, 0, stream>>>(
        buf0, W[3], Bb[3], buf1, 16384, 8192, 512);
    // fused VQ (async-pipelined fp32 WMMA GEMM + argmax + gather)
    vq_kernel<<<VQ_M / 128, 256, 0, stream>>>(
        buf1, cb, enorm, out + (size_t)e * (size_t)VQ_M * VQ_D);
  }
  (void)in_sizes; (void)n_in; (void)out_size; (void)ws_size;
}